// GCN_28252294873753
// MI455X (gfx1250) — compile-verified
//
#include <hip/hip_runtime.h>
#include <math.h>

#define D 128
#define DQ 32   // D/4 float4 chunks per node

typedef float v2f __attribute__((ext_vector_type(2)));
typedef float v8f __attribute__((ext_vector_type(8)));

// ---------------------------------------------------------------------------
// Degree / normalization kernels
// ---------------------------------------------------------------------------
__global__ void k_init_deg(float* __restrict__ deg, int N) {
    int i = blockIdx.x * blockDim.x + threadIdx.x;
    if (i < N) deg[i] = 1.0f;                      // self-loop contribution
}

__global__ void k_accum_deg(const int* __restrict__ dst, float* __restrict__ deg, int E) {
    int e = blockIdx.x * blockDim.x + threadIdx.x;
    if (e < E) unsafeAtomicAdd(&deg[dst[e]], 1.0f); // hw global_atomic_add_f32
}

__global__ void k_dis(float* __restrict__ deg, int N) {
    int i = blockIdx.x * blockDim.x + threadIdx.x;
    if (i < N) deg[i] = rsqrtf(deg[i]);            // deg >= 1 always
}

__global__ void k_norm(const int* __restrict__ src, const int* __restrict__ dst,
                       const float* __restrict__ dis, float* __restrict__ norm, int E) {
    int e = blockIdx.x * blockDim.x + threadIdx.x;
    if (e < E) norm[e] = dis[src[e]] * dis[dst[e]];
}

// ---------------------------------------------------------------------------
// GEMM: H[n][o] = sum_k X[n][k] * W[o][k] + b[o]   (X:[N,128], W:[128,128] row-major)
// One 256-thread block = 8 waves; each wave owns a 16x16 fp32 C tile
// (16 nodes x 16 output features), K-loop via V_WMMA_F32_16X16X4_F32.
// ---------------------------------------------------------------------------
__global__ __launch_bounds__(256) void k_gemm(const float* __restrict__ X,
                                              const float* __restrict__ W,
                                              const float* __restrict__ bias,
                                              float* __restrict__ H, int N) {
    __shared__ float xs[16 * 129];                 // +1 pad: conflict-free LDS
    const int base = blockIdx.x * 16;

    for (int i = threadIdx.x; i < 16 * D; i += 256) {
        int r = i >> 7, c = i & 127;
        int gr = base + r;
        xs[r * 129 + c] = (gr < N) ? X[(size_t)gr * D + c] : 0.0f;
    }
    __syncthreads();

    const int lane  = threadIdx.x & 31;
    const int wave  = threadIdx.x >> 5;
    const int o0    = wave * 16;                   // this wave's output-column tile
    const int mrow  = lane & 15;                   // A: M index / B: N index
    const int khalf = (lane >> 4) * 2;             // lanes 16-31 hold K+2,K+3

    v8f c;
    #pragma unroll
    for (int i = 0; i < 8; ++i) c[i] = 0.0f;

    #pragma unroll
    for (int k0 = 0; k0 < D; k0 += 4) {
        v2f a, b;
        // A fragment: A[M=mrow][k0+khalf .. +1] from LDS
        a.x = xs[mrow * 129 + k0 + khalf];
        a.y = xs[mrow * 129 + k0 + khalf + 1];
        // B fragment: B[k][n] = W[o0+n][k]  (B = W^T slice)
        const int col = o0 + mrow;
        b.x = W[col * D + k0 + khalf];
        b.y = W[col * D + k0 + khalf + 1];
        c = __builtin_amdgcn_wmma_f32_16x16x4_f32(false, a, false, b,
                                                  (short)0, c, false, false);
    }

    // C/D layout: vgpr r -> M=r (lanes 0-15), M=r+8 (lanes 16-31); N = lane&15
    const int col   = o0 + mrow;
    const float bv  = bias[col];
    const int rbase = (lane >> 4) * 8;
    #pragma unroll
    for (int r = 0; r < 8; ++r) {
        int row = base + rbase + r;
        if (row < N) H[(size_t)row * D + col] = c[r] + bv;
    }
}

// ---------------------------------------------------------------------------
// out[i][:] = H[i][:] * dis[i]^2   (self-loop term; also serves as zero-init)
// ---------------------------------------------------------------------------
__global__ void k_selfloop(const float* __restrict__ H, const float* __restrict__ dis,
                           float* __restrict__ out, int N) {
    int idx = blockIdx.x * blockDim.x + threadIdx.x;     // over N*DQ float4s
    if (idx >= N * DQ) return;
    int node = idx >> 5;
    float s = dis[node]; s = s * s;
    float4 v = ((const float4*)H)[idx];
    float4 r; r.x = v.x * s; r.y = v.y * s; r.z = v.z * s; r.w = v.w * s;
    ((float4*)out)[idx] = r;
}

// ---------------------------------------------------------------------------
// Edge scatter: one wave32 per edge, float4 per lane (coalesced 512B gather),
// 4 hardware f32 atomics per lane into out[dst].
// ---------------------------------------------------------------------------
__global__ __launch_bounds__(256) void k_scatter(const float* __restrict__ H,
                                                 const int* __restrict__ src,
                                                 const int* __restrict__ dst,
                                                 const float* __restrict__ norm,
                                                 float* __restrict__ out, int E) {
    int gid  = blockIdx.x * 256 + threadIdx.x;
    int e    = gid >> 5;
    int lane = gid & 31;
    if (e >= E) return;
    int   s = src[e];
    int   d = dst[e];
    float w = norm[e];
    float4 v = ((const float4*)(H + (size_t)s * D))[lane];
    float* o = out + (size_t)d * D + lane * 4;
    unsafeAtomicAdd(o + 0, v.x * w);
    unsafeAtomicAdd(o + 1, v.y * w);
    unsafeAtomicAdd(o + 2, v.z * w);
    unsafeAtomicAdd(o + 3, v.w * w);
}

// ---------------------------------------------------------------------------
// PReLU in place: y = y > 0 ? y : a*y
// ---------------------------------------------------------------------------
__global__ void k_prelu(float* __restrict__ y, const float* __restrict__ a, int total4) {
    int idx = blockIdx.x * blockDim.x + threadIdx.x;
    if (idx >= total4) return;
    float al = a[0];
    float4 v = ((float4*)y)[idx];
    v.x = v.x > 0.0f ? v.x : al * v.x;
    v.y = v.y > 0.0f ? v.y : al * v.y;
    v.z = v.z > 0.0f ? v.z : al * v.z;
    v.w = v.w > 0.0f ? v.w : al * v.w;
    ((float4*)y)[idx] = v;
}

// ---------------------------------------------------------------------------
extern "C" void kernel_launch(void* const* d_in, const int* in_sizes, int n_in,
                              void* d_out, int out_size, void* d_ws, size_t ws_size,
                              hipStream_t stream) {
    const float* x   = (const float*)d_in[0];
    const int*   ei  = (const int*)  d_in[1];
    const float* W1  = (const float*)d_in[2];
    const float* b1  = (const float*)d_in[3];
    const float* a1  = (const float*)d_in[4];
    const float* W2  = (const float*)d_in[5];
    const float* b2  = (const float*)d_in[6];
    const float* a2  = (const float*)d_in[7];
    float* out = (float*)d_out;

    const int N = in_sizes[0] / D;
    const int E = in_sizes[1] / 2;
    const int* src = ei;
    const int* dst = ei + E;

    // workspace layout (floats): dis[N] | norm[E] | h[N*D] | agg[N*D]
    float* dis  = (float*)d_ws;
    float* norm = dis + N;
    float* h    = norm + E;
    float* agg  = h + (size_t)N * D;

    const int T = 256;
    const int gN  = (N + T - 1) / T;
    const int gE  = (E + T - 1) / T;
    const int gND = (N * DQ + T - 1) / T;          // N*D/4 float4s
    const int gEw = (E * 32 + T - 1) / T;          // wave per edge
    const int gM  = (N + 15) / 16;                 // gemm tiles

    // normalization
    k_init_deg <<<gN, T, 0, stream>>>(dis, N);
    k_accum_deg<<<gE, T, 0, stream>>>(dst, dis, E);
    k_dis      <<<gN, T, 0, stream>>>(dis, N);
    k_norm     <<<gE, T, 0, stream>>>(src, dst, dis, norm, E);

    // layer 1: GCNConv + PReLU
    k_gemm     <<<gM,  T, 0, stream>>>(x, W1, b1, h, N);
    k_selfloop <<<gND, T, 0, stream>>>(h, dis, agg, N);
    k_scatter  <<<gEw, T, 0, stream>>>(h, src, dst, norm, agg, E);
    k_prelu    <<<gND, T, 0, stream>>>(agg, a1, N * DQ);

    // layer 2: GCNConv + PReLU -> d_out
    k_gemm     <<<gM,  T, 0, stream>>>(agg, W2, b2, h, N);
    k_selfloop <<<gND, T, 0, stream>>>(h, dis, out, N);
    k_scatter  <<<gEw, T, 0, stream>>>(h, src, dst, norm, out, E);
    k_prelu    <<<gND, T, 0, stream>>>(out, a2, N * DQ);
}